// EdgeDecoder_lgcn_88682484727936
// MI455X (gfx1250) — compile-verified
//
#include <hip/hip_runtime.h>
#include <hip/hip_bf16.h>
#include <math.h>

#define H 128
#define ZPAD 132   // 128 + 4 : bank-conflict-free fragment reads (bank = (4*row + k) % 64)
#define WPAD 68    // 64 + 4  : same trick for the 16x64 weight tile

typedef float v2f __attribute__((ext_vector_type(2)));
typedef float v8f __attribute__((ext_vector_type(8)));

__device__ __forceinline__ v8f wmma_f32(v2f a, v2f b, v8f c) {
    return __builtin_amdgcn_wmma_f32_16x16x4_f32(false, a, false, b, (short)0, c, false, false);
}

// ---------------- Stage 0: one-shot 128x128 transpose of W1 into workspace ----------------
__global__ __launch_bounds__(256) void k_transpose_w1(const float* __restrict__ W1,
                                                      float* __restrict__ W1T)
{
    const int idx = blockIdx.x * 256 + threadIdx.x;
    if (idx < H * H) {
        const int k = idx / H, n = idx % H;       // coalesced read of W1[k][n]
        W1T[n * H + k] = W1[idx];
    }
}

// ---------------- Stage 1 (WMMA): logit[i] = tanh(z_src[i] @ W1 + b1) . w2 ----------------
// Block: 128 threads = 4 waves; block covers 64 z rows, wave w owns rows [blk*64 + w*16, +16).
// A = z rows (M=16 x K=128), B = W1 (K x N=128) staged from W1T so frags are contiguous b64.
__global__ __launch_bounds__(128) void k_logit_wmma(const float* __restrict__ z_src,
                                                    const float* __restrict__ W1T,
                                                    const float* __restrict__ b1,
                                                    const float* __restrict__ w2,
                                                    float* __restrict__ logit, int NS)
{
    __shared__ float W1Ts[H * ZPAD];    // 67.5 KB : [n][k], padded
    __shared__ float zS[64 * ZPAD];     // 33.8 KB : [row][k], padded
    const int tid = threadIdx.x;

    // Stage W1T (rows = W1 columns, contiguous) with float4 copies
    for (int f = tid; f < H * (H / 4); f += 128) {
        const int row = f >> 5;                  // 0..127
        const int cg = (f & 31) * 4;             // k offset
        *(float4*)(W1Ts + row * ZPAD + cg) = *(const float4*)(W1T + (size_t)row * H + cg);
    }
    {
        const int rloc = tid >> 5;
        const int cg = (tid & 31) * 4;
        for (int rr = rloc; rr < 64; rr += 4) {
            int row = blockIdx.x * 64 + rr; if (row >= NS) row = NS - 1;
            *(float4*)(zS + rr * ZPAD + cg) = *(const float4*)(z_src + (size_t)row * H + cg);
        }
    }
    __syncthreads();

    const int wave = tid >> 5;
    const int lane = tid & 31;
    const int l16 = lane & 15;
    const int khalf = lane >> 4;

    const float* aptr = zS + (wave * 16 + l16) * ZPAD + 2 * khalf;
    const float* bbase = W1Ts + l16 * ZPAD + 2 * khalf;   // + nt*16*ZPAD + k0

    v8f acc[8];
    #pragma unroll
    for (int nt = 0; nt < 8; ++nt) acc[nt] = (v8f){};

    for (int k0 = 0; k0 < H; k0 += 4) {
        const v2f a = *(const v2f*)(aptr + k0);
        #pragma unroll
        for (int nt = 0; nt < 8; ++nt) {
            const v2f b = *(const v2f*)(bbase + nt * 16 * ZPAD + k0);
            acc[nt] = wmma_f32(a, b, acc[nt]);
        }
    }

    // Epilogue: rowsum[r] = sum_cols tanh(acc + b1) * w2, cols split over 16 lanes of each half
    float rowsum[8];
    #pragma unroll
    for (int r = 0; r < 8; ++r) rowsum[r] = 0.0f;
    #pragma unroll
    for (int nt = 0; nt < 8; ++nt) {
        const int col = nt * 16 + l16;
        const float bb = b1[col];
        const float ww = w2[col];
        #pragma unroll
        for (int r = 0; r < 8; ++r)
            rowsum[r] += tanhf(acc[nt][r] + bb) * ww;
    }
    #pragma unroll
    for (int m = 1; m <= 8; m <<= 1) {
        #pragma unroll
        for (int r = 0; r < 8; ++r)
            rowsum[r] += __shfl_xor(rowsum[r], m, 32);
    }
    if (l16 == 0) {   // lane 0 (rows 0..7 of tile) and lane 16 (rows 8..15)
        const int rowbase = blockIdx.x * 64 + wave * 16 + 8 * khalf;
        #pragma unroll
        for (int r = 0; r < 8; ++r) {
            const int row = rowbase + r;
            if (row < NS) logit[row] = rowsum[r];
        }
    }
}

// ---------------- Stage 2: per-group masked softmax stats (max m, denom s) ----------------
__global__ __launch_bounds__(256) void k_softmax_stats(const float* __restrict__ logit,
                                                       const int* __restrict__ mask,
                                                       float* __restrict__ mbuf,
                                                       float* __restrict__ sbuf, int NS)
{
    __shared__ float sm[256];
    __shared__ float ss[256];
    const int g = blockIdx.x;
    const int* mrow = mask + (size_t)g * NS;
    float m = -1e30f, s = 0.0f;
    for (int i = threadIdx.x; i < NS; i += 256) {
        if (mrow[i]) {
            const float l = logit[i];
            if (l > m) { s = s * __expf(m - l) + 1.0f; m = l; }
            else       { s += __expf(l - m); }
        }
    }
    sm[threadIdx.x] = m; ss[threadIdx.x] = s;
    __syncthreads();
    for (int off = 128; off > 0; off >>= 1) {
        if (threadIdx.x < off) {
            const float m1 = sm[threadIdx.x], s1 = ss[threadIdx.x];
            const float m2 = sm[threadIdx.x + off], s2 = ss[threadIdx.x + off];
            const float mn = fmaxf(m1, m2);
            sm[threadIdx.x] = mn;
            ss[threadIdx.x] = s1 * __expf(m1 - mn) + s2 * __expf(m2 - mn);
        }
        __syncthreads();
    }
    if (threadIdx.x == 0) { mbuf[g] = sm[0]; sbuf[g] = ss[0]; }
}

// ---------------- Stage 3a (WMMA): partial agg = W(16 x Kchunk) @ z_src(Kchunk x 128) ----------------
// grid = (KC, MT). Block: 128 threads = 4 waves; wave w owns n-tiles {2w, 2w+1}.
__global__ __launch_bounds__(128) void k_agg_wmma(const float* __restrict__ z_src,
                                                  const float* __restrict__ logit,
                                                  const int* __restrict__ mask,
                                                  const float* __restrict__ mbuf,
                                                  float* __restrict__ part,
                                                  int NS, int G, int KC)
{
    __shared__ float zS[64 * ZPAD];   // 33 KB : z rows [k][n]
    __shared__ float wS[16 * WPAD];   //  4 KB : weights [group][k]
    const int tid = threadIdx.x;
    const int wave = tid >> 5;
    const int lane = tid & 31;
    const int l16 = lane & 15;
    const int khalf = lane >> 4;

    const int mt = blockIdx.y;
    const int kc = blockIdx.x;
    const int mbase = mt * 16;
    const int chunk = (NS + KC - 1) / KC;
    const int i0 = kc * chunk;
    const int iend = min(i0 + chunk, NS);

    v8f acc0 = {}, acc1 = {};
    const int c0 = wave * 32 + l16;      // n-tile 2w
    const int c1 = c0 + 16;              // n-tile 2w+1

    for (int base = i0; base < iend; base += 64) {
        {   // stage 64 z_src rows (row-clamped; OOB rows get weight 0 anyway)
            const int rloc = tid >> 5;
            const int cg = (tid & 31) * 4;
            for (int rr = rloc; rr < 64; rr += 4) {
                int row = base + rr; if (row >= NS) row = NS - 1;
                *(float4*)(zS + rr * ZPAD + cg) = *(const float4*)(z_src + (size_t)row * H + cg);
            }
        }
        // stage weights: w[gl][il] = mask ? exp(logit - m_g) : 0
        for (int f = tid; f < 16 * 64; f += 128) {
            const int gl = f >> 6;
            const int il = f & 63;
            const int i = base + il;
            const int g = mbase + gl;
            float w = 0.0f;
            if (i < iend && g < G) {
                if (mask[(size_t)g * NS + i]) w = __expf(logit[i] - mbuf[g]);
            }
            wS[gl * WPAD + il] = w;
        }
        __syncthreads();

        const float* aptr = wS + l16 * WPAD + 2 * khalf;
        #pragma unroll
        for (int k0 = 0; k0 < 64; k0 += 4) {
            const v2f a = *(const v2f*)(aptr + k0);
            const int kr = k0 + 2 * khalf;
            v2f b0, b1v;
            b0.x  = zS[kr * ZPAD + c0];  b0.y  = zS[(kr + 1) * ZPAD + c0];
            b1v.x = zS[kr * ZPAD + c1];  b1v.y = zS[(kr + 1) * ZPAD + c1];
            acc0 = wmma_f32(a, b0, acc0);
            acc1 = wmma_f32(a, b1v, acc1);
        }
        __syncthreads();
    }

    const size_t pb = ((size_t)mt * KC + kc) * 16 * H;
    #pragma unroll
    for (int r = 0; r < 8; ++r) {
        const int row = r + 8 * khalf;
        part[pb + row * H + c0] = acc0[r];
        part[pb + row * H + c1] = acc1[r];
    }
}

// ---------------- Stage 3b: reduce K-chunk partials, scale by 1/s_g ----------------
__global__ __launch_bounds__(128) void k_agg_reduce(const float* __restrict__ part,
                                                    const float* __restrict__ sbuf,
                                                    float* __restrict__ agg, int KC)
{
    const int g = blockIdx.x;
    const int j = threadIdx.x;
    const int mt = g >> 4, gl = g & 15;
    const float s = sbuf[g];
    const float inv = (s > 0.0f) ? (1.0f / s) : 0.0f;
    float acc = 0.0f;
    for (int kc = 0; kc < KC; ++kc)
        acc += part[(((size_t)mt * KC + kc) * 16 + gl) * H + j];
    agg[g * H + j] = acc * inv;
}

// ---------------- Stage 4 (WMMA): scores = agg @ z_dst^T ----------------
__global__ __launch_bounds__(128) void k_scores_wmma(const float* __restrict__ agg,
                                                     const float* __restrict__ z_dst,
                                                     float* __restrict__ out,
                                                     int ND, int G)
{
    __shared__ float aggS[16 * ZPAD];   // A strip
    __shared__ float zdS[64 * ZPAD];    // 64 z_dst rows (= B columns)
    const int tid = threadIdx.x;
    const int mbase = blockIdx.y * 16;
    const int nbase64 = blockIdx.x * 64;

    {
        const int rloc = tid >> 5;
        const int cg = (tid & 31) * 4;
        #pragma unroll
        for (int rr = rloc; rr < 16; rr += 4) {
            int row = mbase + rr; if (row >= G) row = G - 1;
            *(float4*)(aggS + rr * ZPAD + cg) = *(const float4*)(agg + (size_t)row * H + cg);
        }
        for (int rr = rloc; rr < 64; rr += 4) {
            int row = nbase64 + rr; if (row >= ND) row = ND - 1;
            *(float4*)(zdS + rr * ZPAD + cg) = *(const float4*)(z_dst + (size_t)row * H + cg);
        }
    }
    __syncthreads();

    const int wave = tid >> 5;
    const int lane = tid & 31;
    const int l16 = lane & 15;
    const int khalf = lane >> 4;

    const float* aptr = aggS + l16 * ZPAD + 2 * khalf;
    const float* bptr = zdS + (wave * 16 + l16) * ZPAD + 2 * khalf;

    v8f c = {};
    #pragma unroll
    for (int k0 = 0; k0 < H; k0 += 4) {
        v2f a = *(const v2f*)(aptr + k0);
        v2f b = *(const v2f*)(bptr + k0);
        c = wmma_f32(a, b, c);
    }

    const int n = nbase64 + wave * 16 + l16;
    if (n < ND) {
        #pragma unroll
        for (int r = 0; r < 8; ++r) {
            const int m = mbase + r + 8 * khalf;
            if (m < G) out[(size_t)m * ND + n] = c[r];
        }
    }
}

extern "C" void kernel_launch(void* const* d_in, const int* in_sizes, int n_in,
                              void* d_out, int out_size, void* d_ws, size_t ws_size,
                              hipStream_t stream) {
    const float* z_src = (const float*)d_in[0];
    const float* z_dst = (const float*)d_in[1];
    const int*   sym   = (const int*)d_in[2];
    const float* W1    = (const float*)d_in[3];
    const float* b1    = (const float*)d_in[4];
    const float* w2    = (const float*)d_in[5];

    const int NS = in_sizes[0] / H;
    const int ND = in_sizes[1] / H;
    const int G  = in_sizes[2] / NS;
    const int MT = (G + 15) / 16;

    float* out   = (float*)d_out;
    float* ws    = (float*)d_ws;
    float* logit = ws;                               // NS
    float* mbuf  = logit + NS;                       // G
    float* sbuf  = mbuf + G;                         // G
    float* agg   = sbuf + G;                         // G*H
    float* w1t   = agg + (size_t)G * H;              // H*H
    float* part  = w1t + (size_t)H * H;              // MT*KC*16*H

    // Pick K-chunk count for stage 3 from available workspace (deterministic in inputs).
    const size_t baseF  = (size_t)NS + 2 * (size_t)G + (size_t)G * H + (size_t)H * H;
    const size_t availF = (ws_size / 4 > baseF) ? (ws_size / 4 - baseF) : 0;
    int KC = (int)(availF / ((size_t)MT * 16 * H));
    if (KC > 64) KC = 64;
    if (KC < 1)  KC = 1;

    k_transpose_w1<<<(H * H + 255) / 256, 256, 0, stream>>>(W1, w1t);
    k_logit_wmma<<<(NS + 63) / 64, 128, 0, stream>>>(z_src, w1t, b1, w2, logit, NS);
    k_softmax_stats<<<G, 256, 0, stream>>>(logit, sym, mbuf, sbuf, NS);
    dim3 gridC(KC, MT);
    k_agg_wmma<<<gridC, 128, 0, stream>>>(z_src, logit, sym, mbuf, part, NS, G, KC);
    k_agg_reduce<<<G, 128, 0, stream>>>(part, sbuf, agg, KC);
    dim3 gridE((ND + 63) / 64, MT);
    k_scores_wmma<<<gridE, 128, 0, stream>>>(agg, z_dst, out, ND, G);
}